// MeanPooling_40845138985511
// MI455X (gfx1250) — compile-verified
//
#include <hip/hip_runtime.h>

typedef __attribute__((ext_vector_type(2))) float v2f;
typedef __attribute__((ext_vector_type(8))) float v8f;

#define SPLITS 16
#define WAVE_FEATS 16
#define BLOCK_THREADS 256
#define FEATS_PER_BLOCK 128   // 8 waves * 16 features
constexpr int kD = 512;       // feature dim fixed by the reference

// ---- prefix scan of int64 lengths -> int offsets[B+1] (B is tiny: 64) ----
__global__ void seg_offsets_kernel(const long long* __restrict__ lengths,
                                   int* __restrict__ offsets, int B) {
  if (threadIdx.x == 0 && blockIdx.x == 0) {
    int acc = 0;
    offsets[0] = 0;
    for (int i = 0; i < B; ++i) {
      acc += (int)lengths[i];
      offsets[i + 1] = acc;
    }
  }
}

// ---- zero the mean-pool output region (atomics accumulate into it) ----
__global__ void zero_kernel(float* __restrict__ out, int n) {
  int i = blockIdx.x * blockDim.x + threadIdx.x;
  if (i < n) out[i] = 0.0f;
}

// ---- WMMA-based segment mean reduction + attn fill ----
// gridDim.x = B * SPLITS (segment x token-split), gridDim.y = D / 128.
// A = all-ones 16x4 f32 => D[m,n] += sum_k B[k,n]: one wmma reduces a
// 4-token x 16-feature tile; sum is invariant to the B K-striping.
// With kD constexpr, all loads in the unrolled body are constant-offset
// off one running pointer (strength-reduced addressing).
__global__ void __launch_bounds__(BLOCK_THREADS)
mean_pool_wmma_kernel(const float* __restrict__ x,
                      const int* __restrict__ offsets,
                      float* __restrict__ out,
                      float* __restrict__ attn) {
  const int bs = blockIdx.x;
  const int b  = bs / SPLITS;
  const int s  = bs % SPLITS;
  const int start = offsets[b];
  const int end   = offsets[b + 1];
  const int len   = end - start;
  if (len <= 0) return;
  const float inv = 1.0f / (float)len;
  const int per = (len + SPLITS - 1) / SPLITS;
  const int t0  = start + s * per;
  const int t1  = (end < t0 + per) ? end : (t0 + per);

  // attn weights: 1/len per token; only feature-chunk 0 blocks write them
  if (blockIdx.y == 0) {
    for (int i = t0 + (int)threadIdx.x; i < t1; i += blockDim.x) attn[i] = inv;
  }
  if (t0 >= t1) return;

  const int wave = threadIdx.x >> 5;
  const int lane = threadIdx.x & 31;
  const int col  = lane & 15;   // feature column within the wave's 16
  const int half = lane >> 4;   // which pair of K rows this half-wave carries
  const int featBase = blockIdx.y * FEATS_PER_BLOCK + wave * WAVE_FEATS;

  // running per-lane pointer; includes the half-wave's 2-token K offset
  const float* pl = x + (size_t)(t0 + 2 * half) * kD + (size_t)(featBase + col);

  float result;
#if __has_builtin(__builtin_amdgcn_wmma_f32_16x16x4_f32)
  v8f c = {};
  v2f a; a.x = 1.0f; a.y = 1.0f;  // all-ones 16x4 A matrix
  int t = t0;
  // main loop: 32 tokens / iter = 8 WMMAs, 16 constant-offset b32 loads
  for (; t + 32 <= t1; t += 32) {
    __builtin_prefetch(pl + 32 * kD, 0, 1);  // speculative: overrun is dropped
    v2f bb[8];
#pragma unroll
    for (int g = 0; g < 8; ++g) {
      bb[g].x = pl[(size_t)(4 * g) * kD];
      bb[g].y = pl[(size_t)(4 * g + 1) * kD];
    }
#pragma unroll
    for (int g = 0; g < 8; ++g)
      c = __builtin_amdgcn_wmma_f32_16x16x4_f32(false, a, false, bb[g],
                                                (short)0, c, false, false);
    pl += (size_t)32 * kD;
  }
  // 4-token tail steps
  for (; t + 4 <= t1; t += 4) {
    v2f b0;
    b0.x = pl[0];
    b0.y = pl[kD];
    c = __builtin_amdgcn_wmma_f32_16x16x4_f32(false, a, false, b0,
                                              (short)0, c, false, false);
    pl += (size_t)4 * kD;
  }
  // <4-token remainder: plain adds (both half-waves load the same value)
  float tail = 0.0f;
  for (; t < t1; ++t)
    tail += x[(size_t)t * kD + (size_t)(featBase + col)];
  result = c[0] + tail;   // every C row holds the full column sum
#else
  float acc = 0.0f;
  for (int t = t0; t < t1; ++t)
    acc += x[(size_t)t * kD + (size_t)(featBase + col)];
  result = acc;
#endif
  // lanes 16-31 duplicate lanes 0-15; one half commits the scaled partial
  if (lane < 16)
    atomicAdd(&out[(size_t)b * kD + (size_t)(featBase + lane)], result * inv);
}

extern "C" void kernel_launch(void* const* d_in, const int* in_sizes, int n_in,
                              void* d_out, int out_size, void* d_ws, size_t ws_size,
                              hipStream_t stream) {
  (void)n_in; (void)out_size; (void)ws_size;
  const float*     x       = (const float*)d_in[0];
  const long long* lengths = (const long long*)d_in[1];

  const int B = in_sizes[1];        // 64 segments
  int*   offsets = (int*)d_ws;      // B+1 ints of scratch
  float* out     = (float*)d_out;               // [B, D]
  float* attn    = out + (size_t)B * (size_t)kD;  // [N, 1] follows flat

  seg_offsets_kernel<<<1, 32, 0, stream>>>(lengths, offsets, B);

  const int outN = B * kD;
  zero_kernel<<<(outN + 255) / 256, 256, 0, stream>>>(out, outN);

  dim3 grid(B * SPLITS, kD / FEATS_PER_BLOCK);  // 1024 x 4 = 4096 blocks
  mean_pool_wmma_kernel<<<grid, BLOCK_THREADS, 0, stream>>>(x, offsets, out, attn);
}